// PotentialNeuralNet_4320737100384
// MI455X (gfx1250) — compile-verified
//
#include <hip/hip_runtime.h>
#include <math.h>

// Problem constants (match reference setup_inputs()).
#define NTA 500000
#define NC  2500
#define NO  64     // descriptor width (K of forward GEMM)
#define HH  83     // hidden width
#define HP  96     // hidden padded to multiple of 16
#define NS  4      // species

#define TILE 16                 // atoms per wave (M dimension of WMMA)
#define WAVES_PER_BLOCK 2       // 64 threads per block
#define NTILES (NTA / TILE)     // 31250, exact

typedef __attribute__((ext_vector_type(2))) float v2f;
typedef __attribute__((ext_vector_type(8))) float v8f;

__global__ void zero_energy_kernel(float* __restrict__ e) {
    int i = blockIdx.x * 256 + threadIdx.x;
    if (i < NC) e[i] = 0.f;
}

// Butterfly sum over each 16-lane half of the wave32; afterwards every lane
// in a half holds the half's sum.
__device__ __forceinline__ float half_reduce16(float x) {
    x += __shfl_xor(x, 1, 32);
    x += __shfl_xor(x, 2, 32);
    x += __shfl_xor(x, 4, 32);
    x += __shfl_xor(x, 8, 32);
    return x;
}

// f32 WMMA wrapper: D = A(16x4) * B(4x16) + C(16x16)
__device__ __forceinline__ v8f wmma4(v2f a, v2f b, v8f c) {
    return __builtin_amdgcn_wmma_f32_16x16x4_f32(false, a, false, b, (short)0, c,
                                                 false, false);
}

// 8-byte LDS fragment load (lowered to ds_load_b64; addresses are 8B-aligned).
__device__ __forceinline__ v2f lds2(const float* p) {
    return *reinterpret_cast<const v2f*>(p);
}

__launch_bounds__(WAVES_PER_BLOCK * 32)
__global__ void pot_kernel(const int*   __restrict__ symbols,
                           const float* __restrict__ positions,
                           const int*   __restrict__ crystalidx,
                           const float* __restrict__ Wd,   // [3, NO]
                           const float* __restrict__ W1,   // [NS, NO, HH]
                           const float* __restrict__ b1,   // [NS, HH]
                           const float* __restrict__ W2,   // [NS, HH, 1]
                           const float* __restrict__ b2,   // [NS, 1]
                           float* __restrict__ energies,   // [NTA]
                           float* __restrict__ energy,     // [NC] (pre-zeroed)
                           float* __restrict__ forces)     // [NTA, 3]
{
    // LDS: 24576 + 384 + 384 + 768 + 8192 + 12288 = 46592 B
    __shared__ __align__(16) float w1L[NO * HP];   // W1[s], zero-padded cols
    __shared__ __align__(16) float b1L[HP];
    __shared__ __align__(16) float w2L[HP];
    __shared__ __align__(16) float wdL[3 * NO];
    __shared__ __align__(16) float descL[WAVES_PER_BLOCK][TILE * NO];
    __shared__ __align__(16) float gL[WAVES_PER_BLOCK][TILE * HP];

    const int tid  = threadIdx.x;
    const int wave = tid >> 5;
    const int lane = tid & 31;
    const int lm   = lane & 15;      // A-frag: M index | B-frag: N index
    const int lh   = lane >> 4;      // lane half
    const int k0   = 2 * lh;         // A/B frag: K offset of this half
    const int moff = 8 * lh;         // D-layout: row offset of this half

    const int tileIdx = blockIdx.x * WAVES_PER_BLOCK + wave;   // always < NTILES
    const int base    = tileIdx * TILE;

    // Stage Wd (shared by desc GEMM and force projection).
    for (int i = tid; i < 3 * NO; i += WAVES_PER_BLOCK * 32) wdL[i] = Wd[i];
    __syncthreads();

    // ---------------- desc = sin(pos @ Wd);  keep cos(z) in registers -------
    // A = positions tile 16x3 zero-padded to 16x4 (K rows 3 -> 4).
    v2f apos;
    apos.x = (k0     < 3) ? positions[(base + lm) * 3 + k0    ] : 0.f;
    apos.y = (k0 + 1 < 3) ? positions[(base + lm) * 3 + k0 + 1] : 0.f;

    v8f cosz[4];
    #pragma unroll
    for (int nt = 0; nt < 4; ++nt) {
        const int n = 16 * nt + lm;
        v2f b;
        b.x = (k0     < 3) ? wdL[(k0    ) * NO + n] : 0.f;
        b.y = (k0 + 1 < 3) ? wdL[(k0 + 1) * NO + n] : 0.f;
        v8f z = {0.f, 0.f, 0.f, 0.f, 0.f, 0.f, 0.f, 0.f};
        z = wmma4(apos, b, z);
        #pragma unroll
        for (int r = 0; r < 8; ++r) {
            float sz, cz;
            __sincosf(z[r], &sz, &cz);
            cosz[nt][r] = cz;
            // D-layout element (m = r+moff, n) -> row-major LDS for A-frags
            descL[wave][(r + moff) * NO + n] = sz;
        }
    }

    // Species of the 8 atom rows this lane's D-layout elements belong to.
    int symv[8];
    #pragma unroll
    for (int r = 0; r < 8; ++r) symv[r] = symbols[base + r + moff];

    v8f   gsel[4];
    float esel[8];
    #pragma unroll
    for (int nt = 0; nt < 4; ++nt) gsel[nt] = (v8f){0,0,0,0,0,0,0,0};
    #pragma unroll
    for (int r = 0; r < 8; ++r) esel[r] = 0.f;

    __syncthreads();   // desc / wdL staged

    // ---------------- per-species MLP fwd + bwd ------------------------------
    for (int s = 0; s < NS; ++s) {
        // Stage padded W1[s] (rows = desc index, cols = hidden), b1[s], W2[s].
        for (int i = tid; i < NO * HP; i += WAVES_PER_BLOCK * 32) {
            const int row = i / HP, col = i % HP;
            w1L[i] = (col < HH) ? W1[(s * NO + row) * HH + col] : 0.f;
        }
        for (int i = tid; i < HP; i += WAVES_PER_BLOCK * 32) {
            b1L[i] = (i < HH) ? b1[s * HH + i] : 0.f;
            w2L[i] = (i < HH) ? W2[s * HH + i] : 0.f;
        }
        const float b2s = b2[s];
        __syncthreads();

        // Forward: u = desc(16x64) @ W1(64x96) + b1, then silu pieces.
        float esacc[8];
        #pragma unroll
        for (int r = 0; r < 8; ++r) esacc[r] = 0.f;

        #pragma unroll
        for (int nt = 0; nt < 6; ++nt) {
            const int n = 16 * nt + lm;
            const float b1v = b1L[n];
            v8f u;
            #pragma unroll
            for (int r = 0; r < 8; ++r) u[r] = b1v;
            #pragma unroll 4
            for (int kb = 0; kb < NO; kb += 4) {
                // A: two consecutive K floats -> single ds_load_b64
                v2f a = lds2(&descL[wave][lm * NO + kb + k0]);
                v2f b;   // W1 column pair (stride HP): two b32 loads
                b.x = w1L[(kb + k0    ) * HP + n];
                b.y = w1L[(kb + k0 + 1) * HP + n];
                u = wmma4(a, b, u);
            }
            const float w2v = w2L[n];
            #pragma unroll
            for (int r = 0; r < 8; ++r) {
                const float uv  = u[r];
                const float sig = 1.f / (1.f + __expf(-uv));
                const float h   = uv * sig;                       // silu
                const float dsl = sig * (1.f + uv * (1.f - sig)); // silu'
                esacc[r] += h * w2v;
                gL[wave][(r + moff) * HP + n] = w2v * dsl;        // dE/du
            }
        }
        __syncthreads();   // gL staged (uniform across block)

        // Backward: dE/ddesc = g_u(16x96) @ W1^T(96x64).
        v8f gd[4];
        #pragma unroll
        for (int nt = 0; nt < 4; ++nt) {
            const int n = 16 * nt + lm;          // desc-feature index
            v8f acc = {0.f, 0.f, 0.f, 0.f, 0.f, 0.f, 0.f, 0.f};
            #pragma unroll 4
            for (int kb = 0; kb < HP; kb += 4) {
                // Both fragments contiguous in K -> ds_load_b64 each
                v2f a = lds2(&gL[wave][lm * HP + kb + k0]);
                v2f b = lds2(&w1L[n * HP + kb + k0]);   // W1^T(k=hid, n=feat)
                acc = wmma4(a, b, acc);
            }
            gd[nt] = acc;
        }

        // Per-atom species selection (mirrors jnp.where in the reference).
        #pragma unroll
        for (int r = 0; r < 8; ++r) {
            const float es = half_reduce16(esacc[r]) + b2s;
            const bool sel = (symv[r] == s);
            if (sel) {
                esel[r] = es;
                #pragma unroll
                for (int nt = 0; nt < 4; ++nt) gsel[nt][r] = gd[nt][r];
            }
        }
        __syncthreads();   // protect w1L before next species staging
    }

    // ---------------- forces, energies, crystal energy -----------------------
    // f[m,k] = sum_n gsel[m,n] * cos(z[m,n]) * Wd[k,n]
    #pragma unroll
    for (int r = 0; r < 8; ++r) {
        float f0 = 0.f, f1 = 0.f, f2 = 0.f;
        #pragma unroll
        for (int nt = 0; nt < 4; ++nt) {
            const int   n  = 16 * nt + lm;
            const float gg = gsel[nt][r] * cosz[nt][r];
            f0 += gg * wdL[0 * NO + n];
            f1 += gg * wdL[1 * NO + n];
            f2 += gg * wdL[2 * NO + n];
        }
        f0 = half_reduce16(f0);
        f1 = half_reduce16(f1);
        f2 = half_reduce16(f2);
        if (lm == r) {   // lanes r and r+16 commit rows r and r+8
            const int atom = base + r + moff;
            forces[atom * 3 + 0] = f0;
            forces[atom * 3 + 1] = f1;
            forces[atom * 3 + 2] = f2;
            energies[atom] = esel[r];
            __hip_atomic_fetch_add(&energy[crystalidx[atom]], esel[r],
                                   __ATOMIC_RELAXED, __HIP_MEMORY_SCOPE_AGENT);
        }
    }
}

extern "C" void kernel_launch(void* const* d_in, const int* in_sizes, int n_in,
                              void* d_out, int out_size, void* d_ws, size_t ws_size,
                              hipStream_t stream) {
    (void)in_sizes; (void)n_in; (void)out_size; (void)d_ws; (void)ws_size;
    const int*   symbols    = (const int*)  d_in[0];
    const float* positions  = (const float*)d_in[1];
    /* d_in[2] = cells (unused by reference math) */
    const int*   crystalidx = (const int*)  d_in[3];
    /* d_in[4] = pbcs (unused) */
    const float* Wd = (const float*)d_in[5];
    const float* W1 = (const float*)d_in[6];
    const float* b1 = (const float*)d_in[7];
    const float* W2 = (const float*)d_in[8];
    const float* b2 = (const float*)d_in[9];

    float* out      = (float*)d_out;
    float* energies = out;               // [NTA]
    float* energy   = out + NTA;         // [NC]
    float* forces   = out + NTA + NC;    // [NTA*3]

    zero_energy_kernel<<<(NC + 255) / 256, 256, 0, stream>>>(energy);

    const int nblocks = NTILES / WAVES_PER_BLOCK;   // 15625, exact
    pot_kernel<<<nblocks, WAVES_PER_BLOCK * 32, 0, stream>>>(
        symbols, positions, crystalidx, Wd, W1, b1, W2, b2,
        energies, energy, forces);
}